// MambaLayer_29145648070726
// MI455X (gfx1250) — compile-verified
//
#include <hip/hip_runtime.h>
#include <hip/hip_bf16.h>
#include <math.h>

typedef float v2f __attribute__((ext_vector_type(2)));
typedef float v8f __attribute__((ext_vector_type(8)));

#define LSEQ 4096           // H*W
#define NB   8              // batch
#define DIN  512            // d_inner
#define NST  16             // d_state
#define DTR  16             // dt_rank
#define CDIM 256            // model dim

// ---------------------------------------------------------------------------
// LayerNorm over channel dim; x is (B, C, L); output xn kept in (B, C, L).
// ---------------------------------------------------------------------------
__global__ __launch_bounds__(256)
void ln_kernel(const float* __restrict__ x, const float* __restrict__ g,
               const float* __restrict__ be, float* __restrict__ xn) {
  int l = blockIdx.x * 256 + threadIdx.x;
  int b = blockIdx.y;
  const float* xb = x + (((size_t)b * CDIM) << 12) + l;
  float s = 0.f, s2 = 0.f;
  for (int c = 0; c < CDIM; ++c) {
    float v = xb[(size_t)c << 12];
    s += v; s2 += v * v;
  }
  float mu = s * (1.0f / CDIM);
  float var = s2 * (1.0f / CDIM) - mu * mu;
  float rs = rsqrtf(var + 1e-5f);
  float* ob = xn + (((size_t)b * CDIM) << 12) + l;
  for (int c = 0; c < CDIM; ++c) {
    float v = xb[(size_t)c << 12];
    ob[(size_t)c << 12] = (v - mu) * rs * g[c] + be[c];
  }
}

// ---------------------------------------------------------------------------
// WMMA f32 GEMM, NT column-tiles per wave (16 x 16*NT output per wave):
//   D[(b,col,l)] = sum_k (A[(b,k,l)] (+A2 if FUSE2)) * W[col][k]
// A layout (Bm, AC, LSEQ) channel-major; W row-major (N x K).
// If W1 != null, row-tiles in the upper half of tiles_m use W1/b1 (merged
// fwd/bwd directions).  epilogue: soft=1 -> softplus(x + bias[col]).
// FUSE2 is compile-time so the non-fused GEMMs get a branch-free K-loop.
// ---------------------------------------------------------------------------
template <int NT, bool FUSE2>
__global__ __launch_bounds__(256)
void gemm_wmma(const float* __restrict__ A, const float* __restrict__ A2,
               int AC, int K,
               const float* __restrict__ W0, const float* __restrict__ W1,
               const float* __restrict__ b0, const float* __restrict__ b1,
               int soft, float* __restrict__ Dout, int tiles_n, int tiles_m) {
  const int groups = tiles_n / NT;
  int wid  = blockIdx.x * 8 + (threadIdx.x >> 5);
  int lane = threadIdx.x & 31;
  int tg = wid % groups;
  int tm = wid / groups;
  if (tm >= tiles_m) return;

  int dir = (W1 != nullptr && tm >= (tiles_m >> 1)) ? 1 : 0;
  const float* W    = dir ? W1 : W0;
  const float* bias = dir ? b1 : b0;

  int row0 = tm << 4;               // global row in (Bm*L)
  int b    = row0 >> 12;            // L = 4096
  int l0   = row0 & 4095;
  int m    = lane & 15;
  int kh   = lane >> 4;
  int col0 = (tg * NT) << 4;

  const float* Arow  = A + (((size_t)b * AC) << 12) + l0 + m;
  const float* A2row = FUSE2 ? (A2 + (((size_t)b * AC) << 12) + l0 + m) : nullptr;
  const float* Wr[NT];
#pragma unroll
  for (int t = 0; t < NT; ++t)
    Wr[t] = W + (size_t)(col0 + t * 16 + m) * K + 2 * kh;

  v8f c[NT];
#pragma unroll
  for (int t = 0; t < NT; ++t) c[t] = (v8f){0.f,0.f,0.f,0.f,0.f,0.f,0.f,0.f};

#pragma unroll 4
  for (int k = 0; k < K; k += 4) {
    v2f af;
    const float* ap = Arow + (((size_t)(k + 2 * kh)) << 12);
    if (k + 16 < K)                     // prefetch A-stream 16 k-steps ahead
      __builtin_prefetch(ap + ((size_t)16 << 12));
    af.x = ap[0];
    af.y = ap[4096];
    if (FUSE2) {                        // fused y_f + y_b for out_proj
      const float* ap2 = A2row + (((size_t)(k + 2 * kh)) << 12);
      af.x += ap2[0];
      af.y += ap2[4096];
    }
#pragma unroll
    for (int t = 0; t < NT; ++t) {
      v2f bf = *(const v2f*)(Wr[t] + k);
      c[t] = __builtin_amdgcn_wmma_f32_16x16x4_f32(
          false, af, false, bf, (short)0, c[t], false, false);
    }
  }

  int NC = tiles_n << 4;
#pragma unroll
  for (int t = 0; t < NT; ++t) {
    int col = col0 + t * 16 + m;
    float bval = bias ? bias[col] : 0.f;
    float vout[8];
#pragma unroll
    for (int j = 0; j < 8; ++j) {
      float v = c[t][j];
      if (soft) {
        v += bval;
        v = (v > 20.f) ? v : log1pf(__expf(v));   // softplus
      }
      vout[j] = v;
    }
    size_t ob = (((size_t)(b * NC + col)) << 12) + l0 + (kh << 3);
    *(float4*)(Dout + ob)     = make_float4(vout[0], vout[1], vout[2], vout[3]);
    *(float4*)(Dout + ob + 4) = make_float4(vout[4], vout[5], vout[6], vout[7]);
  }
}

// ---------------------------------------------------------------------------
// Depthwise causal conv (k=4) + SiLU, both directions in one launch.
// Input = x-half of xz (B,1024,L); output xc (2,B,512,L) in logical order.
// ---------------------------------------------------------------------------
__global__ __launch_bounds__(256)
void conv_silu(const float* __restrict__ xz,
               const float* __restrict__ w0, const float* __restrict__ cb0,
               const float* __restrict__ w1, const float* __restrict__ cb1,
               float* __restrict__ xc) {
  int l   = blockIdx.x * 256 + threadIdx.x;   // logical position
  int d   = blockIdx.y;
  int bz  = blockIdx.z;                       // 0..15
  int b   = bz & 7;
  int rev = bz >> 3;
  const float* w  = rev ? w1 : w0;
  const float* cb = rev ? cb1 : cb0;
  const float* xp = xz + (((size_t)(b * 2 * DIN + d)) << 12);
  float acc = cb[d];
#pragma unroll
  for (int j = 0; j < 4; ++j) {
    int ll = l - 3 + j;
    if (ll >= 0) {
      int lp = rev ? (LSEQ - 1 - ll) : ll;
      acc = fmaf(w[d * 4 + j], xp[lp], acc);
    }
  }
  float sv = acc / (1.f + __expf(-acc));      // SiLU
  xc[(((size_t)(bz * DIN + d)) << 12) + l] = sv;
}

// ---------------------------------------------------------------------------
// Selective scan, lane-parallel over the state dim: one lane per (dir,b,d,n).
// Both directions concurrent (blockIdx.z = dir).  4 timesteps per iteration
// with float4 streaming loads; sum_n h*C reduced via wave32 __shfl_xor.
// Gated output written as aligned float4 to the direction's y buffer at the
// physical (un-reversed) position.
// ---------------------------------------------------------------------------
__global__ __launch_bounds__(512)
void scan_kernel(const float* __restrict__ delta, const float* __restrict__ xc,
                 const float* __restrict__ dbl,   const float* __restrict__ xz,
                 const float* __restrict__ A_log0, const float* __restrict__ Dp0,
                 const float* __restrict__ A_log1, const float* __restrict__ Dp1,
                 float* __restrict__ ybuf) {
  int tid  = threadIdx.x;
  int lane = tid & 31;
  int n    = lane & 15;
  int dg   = tid >> 4;
  int d    = blockIdx.x * 32 + dg;
  int b    = blockIdx.y;
  int rev  = blockIdx.z;
  int bz   = rev * 8 + b;                      // merged (dir,b) index

  const float* A_log = rev ? A_log1 : A_log0;
  const float* Dp    = rev ? Dp1    : Dp0;
  float An = -__expf(A_log[d * NST + n]);
  float Dd = Dp[d];

  const float* dP = delta + (((size_t)(bz * DIN + d)) << 12);
  const float* uP = xc    + (((size_t)(bz * DIN + d)) << 12);
  const float* BP = dbl   + (((size_t)(bz * 48 + DTR + n)) << 12);
  const float* CP = dbl   + (((size_t)(bz * 48 + DTR + NST + n)) << 12);
  const float* zP = xz    + (((size_t)(b * 2 * DIN + DIN + d)) << 12);
  float*       yP = ybuf  + (((size_t)(bz * DIN + d)) << 12);

  float h = 0.f;
  for (int l0 = 0; l0 < LSEQ; l0 += 4) {
    float4 dt4 = *(const float4*)(dP + l0);
    float4 u4  = *(const float4*)(uP + l0);
    float4 B4  = *(const float4*)(BP + l0);
    float4 C4  = *(const float4*)(CP + l0);
    const float* dtv = (const float*)&dt4;
    const float* uv  = (const float*)&u4;
    const float* Bv  = (const float*)&B4;
    const float* Cv  = (const float*)&C4;
    float ys[4];
#pragma unroll
    for (int j = 0; j < 4; ++j) {
      float a = __expf(dtv[j] * An);
      h = fmaf(a, h, dtv[j] * uv[j] * Bv[j]);   // h = exp(dt*A)*h + (dt*u)*B
      float p = h * Cv[j];
      p += __shfl_xor(p, 1, 16);                // reduce over 16 states
      p += __shfl_xor(p, 2, 16);
      p += __shfl_xor(p, 4, 16);
      p += __shfl_xor(p, 8, 16);
      ys[j] = p;
    }
    if (n == 0) {
      int base = rev ? (LSEQ - 4 - l0) : l0;    // aligned physical group
      float4 z4 = *(const float4*)(zP + base);
      const float* zv = (const float*)&z4;
      float o[4];
#pragma unroll
      for (int j = 0; j < 4; ++j) {
        int jj = rev ? (3 - j) : j;             // position within the group
        float z = zv[jj];
        o[jj] = (ys[j] + uv[j] * Dd) * (z / (1.f + __expf(-z)));
      }
      *(float4*)(yP + base) = make_float4(o[0], o[1], o[2], o[3]);
    }
  }
}

// ---------------------------------------------------------------------------
// Host-side orchestration
// ---------------------------------------------------------------------------
extern "C" void kernel_launch(void* const* d_in, const int* in_sizes, int n_in,
                              void* d_out, int out_size, void* d_ws, size_t ws_size,
                              hipStream_t stream) {
  const float* x      = (const float*)d_in[0];
  const float* ln_w   = (const float*)d_in[1];
  const float* ln_b   = (const float*)d_in[2];
  const float* in_w   = (const float*)d_in[3];   // (1024,256)
  const float* cw0    = (const float*)d_in[4];
  const float* cb0    = (const float*)d_in[5];
  const float* xpw0   = (const float*)d_in[6];
  const float* dtw0   = (const float*)d_in[7];
  const float* dtb0   = (const float*)d_in[8];
  const float* alog0  = (const float*)d_in[9];
  const float* Dv0    = (const float*)d_in[10];
  const float* cw1    = (const float*)d_in[11];
  const float* cb1    = (const float*)d_in[12];
  const float* xpw1   = (const float*)d_in[13];
  const float* dtw1   = (const float*)d_in[14];
  const float* dtb1   = (const float*)d_in[15];
  const float* alog1  = (const float*)d_in[16];
  const float* Dv1    = (const float*)d_in[17];
  const float* out_w  = (const float*)d_in[18];  // (256,512)
  float* out = (float*)d_out;

  // workspace layout (floats); activations in (b, chan, L) layout,
  // dual-direction buffers are (2, B, chan, L)
  float* ws    = (float*)d_ws;
  float* xn    = ws;                                       // 8*256*4096
  float* xz    = xn    + (size_t)NB * CDIM * LSEQ;         // 8*1024*4096
  float* xc2   = xz    + (size_t)NB * 2 * DIN * LSEQ;      // 2*8*512*4096
  float* dbl2  = xc2   + (size_t)2 * NB * DIN * LSEQ;      // 2*8*48*4096
  float* dlt2  = dbl2  + (size_t)2 * NB * 48 * LSEQ;       // 2*8*512*4096
  float* y2    = dlt2  + (size_t)2 * NB * DIN * LSEQ;      // 2*8*512*4096
  float* y1    = y2    + (size_t)NB * DIN * LSEQ;          // second dir half

  const int TM1 = (NB * LSEQ) / 16;        // 2048 (single set of batches)
  const int TM2 = 2 * TM1;                 // 4096 (both directions merged)

  // 1) LayerNorm
  ln_kernel<<<dim3(LSEQ / 256, NB), 256, 0, stream>>>(x, ln_w, ln_b, xn);

  // 2) in_proj: xz(b,1024,l) = xn @ in_w^T          K=256, N=1024
  gemm_wmma<4, false><<<(TM1 * (1024 / 64)), 256, 0, stream>>>(
      xn, nullptr, CDIM, CDIM, in_w, nullptr, nullptr, nullptr, 0,
      xz, 1024 / 16, TM1);

  // 3) conv+SiLU, both directions in one launch
  conv_silu<<<dim3(LSEQ / 256, DIN, 2 * NB), 256, 0, stream>>>(
      xz, cw0, cb0, cw1, cb1, xc2);

  // 4) x_proj (merged dirs): dbl(2b,48,l) = xc @ xpw^T   K=512, N=48
  gemm_wmma<3, false><<<(TM2 * 1) / 8, 256, 0, stream>>>(
      xc2, nullptr, DIN, DIN, xpw0, xpw1, nullptr, nullptr, 0,
      dbl2, 3, TM2);

  // 5) dt_proj + softplus (merged dirs): K=16, N=512
  gemm_wmma<4, false><<<(TM2 * (DIN / 64)), 256, 0, stream>>>(
      dbl2, nullptr, 48, DTR, dtw0, dtw1, dtb0, dtb1, 1,
      dlt2, DIN / 16, TM2);

  // 6) selective scan + gating, both directions concurrent
  scan_kernel<<<dim3(DIN / 32, NB, 2), 512, 0, stream>>>(
      dlt2, xc2, dbl2, xz, alog0, Dv0, alog1, Dv1, y2);

  // 7) out_proj with fused y_f + y_b, straight into d_out (B,256,H,W)
  gemm_wmma<4, true><<<(TM1 * (CDIM / 64)), 256, 0, stream>>>(
      y2, y1, DIN, DIN, out_w, nullptr, nullptr, nullptr, 0,
      out, CDIM / 16, TM1);
}